// SofaNetEllipse_54546084660107
// MI455X (gfx1250) — compile-verified
//
#include <hip/hip_runtime.h>

// SofaNet ellipse kernel for gfx1250 (MI455X), wave32 + f32 WMMA 16x16x4.

typedef float v2f __attribute__((ext_vector_type(2)));
typedef float v4f __attribute__((ext_vector_type(4)));
typedef float v8f __attribute__((ext_vector_type(8)));

#define N_NET 128
#define H 128
#define L_FULL 4097
#define BHALF 2049              // L/2 + 1
#define COLS 64                 // batch columns per workgroup
#define NL (N_NET * L_FULL)

// One WMMA MLP layer: Y = relu(W @ X + b), TY = (pre>0) ? (W @ T) : 0
// X/T/Xout/Tout are LDS, batch-major: buf[col*H + h], col in [0,COLS).
// Each wave handles 16 columns (group g) and 4 of the 8 row-tiles (mtHalf).
__device__ __forceinline__ void wmma_layer(
    const float* __restrict__ Wg,   // [H][H] row-major (global, L2-resident)
    const float* __restrict__ Bg,   // [H] bias (global)
    const float* Xin, const float* Tin,
    float* Xout, float* Tout,
    int lane, int g, int mtHalf)
{
    const int hi  = lane >> 4;          // half-wave: 0 or 1
    const int lo  = lane & 15;
    const int col = g * 16 + lo;        // batch column owned by this lane (B/C/D N-index)

    #pragma unroll
    for (int mi = 0; mi < 4; ++mi) {
        const int mt = (mtHalf << 2) + mi;          // 16-row output tile
        v8f cp = {0.f,0.f,0.f,0.f,0.f,0.f,0.f,0.f}; // primal accumulator
        v8f ct = {0.f,0.f,0.f,0.f,0.f,0.f,0.f,0.f}; // tangent accumulator
        const float* wrow = Wg + (mt * 16 + lo) * H; // A-matrix row for this lane

        #pragma unroll 8
        for (int k0 = 0; k0 < H; k0 += 4) {
            const int kk = k0 + hi * 2;  // A/B frag: lane holds K = kk, kk+1
            v2f af = *(const v2f*)(wrow + kk);            // global b64
            v2f bp = *(const v2f*)(Xin + col * H + kk);   // ds b64
            v2f bt = *(const v2f*)(Tin + col * H + kk);   // ds b64
            cp = __builtin_amdgcn_wmma_f32_16x16x4_f32(
                     false, af, false, bp, (short)0, cp, false, false);
            ct = __builtin_amdgcn_wmma_f32_16x16x4_f32(
                     false, af, false, bt, (short)0, ct, false, false);
        }

        // C/D layout: VGPR v <-> row M = mt*16 + hi*8 + v, N = col.
        const int m0 = mt * 16 + hi * 8;
        v4f bb0 = *(const v4f*)(Bg + m0);
        v4f bb1 = *(const v4f*)(Bg + m0 + 4);
        v4f x0, x1, t0, t1;
        #pragma unroll
        for (int v = 0; v < 4; ++v) {
            float pre = cp[v] + bb0[v];
            x0[v] = pre > 0.f ? pre   : 0.f;
            t0[v] = pre > 0.f ? ct[v] : 0.f;
        }
        #pragma unroll
        for (int v = 0; v < 4; ++v) {
            float pre = cp[v + 4] + bb1[v];
            x1[v] = pre > 0.f ? pre       : 0.f;
            t1[v] = pre > 0.f ? ct[v + 4] : 0.f;
        }
        *(v4f*)(Xout + col * H + m0)     = x0;   // ds b128
        *(v4f*)(Xout + col * H + m0 + 4) = x1;
        *(v4f*)(Tout + col * H + m0)     = t0;
        *(v4f*)(Tout + col * H + m0 + 4) = t1;
    }
}

__global__ __launch_bounds__(256)
void SofaNetEllipse_kernel(
    const float* __restrict__ alpha,
    const float* __restrict__ w0, const float* __restrict__ w1,
    const float* __restrict__ w2, const float* __restrict__ w3,
    const float* __restrict__ b0, const float* __restrict__ b1,
    const float* __restrict__ b2, const float* __restrict__ b3,
    const float* __restrict__ sqrt_a,
    float* __restrict__ out)
{
    __shared__ float XA[COLS * H];  // 32 KB each; 128 KB total (<320 KB/WGP)
    __shared__ float TA[COLS * H];
    __shared__ float XB[COLS * H];
    __shared__ float TB[COLS * H];

    const int n    = blockIdx.x;           // network id
    const int j0   = blockIdx.y * COLS;    // first batch column of this tile
    const int tid  = threadIdx.x;
    const int lane = tid & 31;
    const int wave = tid >> 5;             // 8 waves

    // ---- Layer 0: x0 = relu(w0*alpha + b0), t0 = (pre>0) ? w0 : 0 ----
    {
        const int h = tid & 127;
        const float w0v = w0[n * H + h];
        const float b0v = b0[n * H + h];
        #pragma unroll
        for (int i = 0; i < COLS / 2; ++i) {
            const int c = (tid >> 7) + 2 * i;  // 2 column-halves x 32 iters covers 64 cols
            int jb = j0 + c;
            if (jb > BHALF - 1) jb = BHALF - 1;           // clamp padded cols
            const float a   = alpha[jb];
            const float pre = w0v * a + b0v;
            XA[c * H + h] = pre > 0.f ? pre : 0.f;
            TA[c * H + h] = pre > 0.f ? w0v : 0.f;
        }
    }
    __syncthreads();

    // ---- Layers 1 & 2 via WMMA (EXEC all-ones: fully uniform control flow) ----
    const int g      = wave & 3;   // column group (16 cols)
    const int mtHalf = wave >> 1 >> 1; // wave>>2: which half of row-tiles
    wmma_layer(w1 + n * H * H, b1 + n * H, XA, TA, XB, TB, lane, g, mtHalf);
    __syncthreads();
    wmma_layer(w2 + n * H * H, b2 + n * H, XB, TB, XA, TA, lane, g, mtHalf);
    __syncthreads();

    // ---- Layer 3 + square + trig outputs (+mirror) ----
    if (tid < COLS) {
        const int c = tid;
        const int j = j0 + c;
        if (j < BHALF) {
            const float* __restrict__ w3n = w3 + n * H;
            float accp = 0.f, acct = 0.f;
            #pragma unroll 4
            for (int k = 0; k < H; ++k) {
                const float wk = w3n[k];
                accp += wk * XA[c * H + k];
                acct += wk * TA[c * H + k];
            }
            const float s   = accp + b3[n];
            const float bv  = s * s;              // b
            const float dbv = 2.f * s * acct;     // db (JVP of square)
            const float an  = sqrt_a[n] * sqrt_a[n];

            {
                const float al = alpha[j];
                const float sn = sinf(al), cs = cosf(al);
                out[0 * NL + n * L_FULL + j] = an * (cs - 1.f);
                out[1 * NL + n * L_FULL + j] = bv * sn;
                out[2 * NL + n * L_FULL + j] = -an * sn;
                out[3 * NL + n * L_FULL + j] = bv * cs + dbv * sn;
            }
            if (j < BHALF - 1) {                   // mirror half: b same, db negated
                const int i2 = (L_FULL - 1) - j;   // 4096 - j
                const float al = alpha[i2];
                const float sn = sinf(al), cs = cosf(al);
                out[0 * NL + n * L_FULL + i2] = an * (cs - 1.f);
                out[1 * NL + n * L_FULL + i2] = bv * sn;
                out[2 * NL + n * L_FULL + i2] = -an * sn;
                out[3 * NL + n * L_FULL + i2] = bv * cs - dbv * sn;
            }
        }
    }
}

extern "C" void kernel_launch(void* const* d_in, const int* in_sizes, int n_in,
                              void* d_out, int out_size, void* d_ws, size_t ws_size,
                              hipStream_t stream) {
    (void)in_sizes; (void)n_in; (void)d_ws; (void)ws_size; (void)out_size;
    // setup_inputs order: alpha, w0, w1, w2, w3, b0, b1, b2, b3, sqrt_a  (all f32)
    const float* alpha  = (const float*)d_in[0];
    const float* w0     = (const float*)d_in[1];
    const float* w1     = (const float*)d_in[2];
    const float* w2     = (const float*)d_in[3];
    const float* w3     = (const float*)d_in[4];
    const float* b0     = (const float*)d_in[5];
    const float* b1     = (const float*)d_in[6];
    const float* b2     = (const float*)d_in[7];
    const float* b3     = (const float*)d_in[8];
    const float* sqrt_a = (const float*)d_in[9];
    float* out = (float*)d_out;

    dim3 grid(N_NET, (BHALF + COLS - 1) / COLS);   // 128 x 33 workgroups
    SofaNetEllipse_kernel<<<grid, 256, 0, stream>>>(
        alpha, w0, w1, w2, w3, b0, b1, b2, b3, sqrt_a, out);
}